// _ProposalLayer_24498493456861
// MI455X (gfx1250) — compile-verified
//
#include <hip/hip_runtime.h>
#include <stdint.h>

#define NB 4
#define NA 9
#define FH 64
#define FW 64
#define NANCH (FH*FW*NA)      // 36864
#define PSORT 65536
#define PRE 6000
#define POST 300
#define MASKW 188             // ceil(6000/32)
#define NMS_T 0.7f

typedef float v2f __attribute__((ext_vector_type(2)));
typedef float v8f __attribute__((ext_vector_type(8)));

// py-faster-rcnn generate_anchors(16, [.5,1,2], [8,16,32]) — precomputed (np.round half-to-even)
__constant__ float kBase[NA][4] = {
  {-84.f,-40.f,99.f,55.f},   {-176.f,-88.f,191.f,103.f}, {-360.f,-184.f,375.f,199.f},
  {-56.f,-56.f,71.f,71.f},   {-120.f,-120.f,135.f,135.f},{-248.f,-248.f,263.f,263.f},
  {-36.f,-80.f,51.f,95.f},   {-80.f,-168.f,95.f,183.f},  {-168.f,-344.f,183.f,359.f}
};

__global__ void k_zero(float* p, int n) {
  int i = blockIdx.x * blockDim.x + threadIdx.x;
  if (i < n) p[i] = 0.0f;
}

// 64-bit sort key: high 32 = bit-flipped float (descending order), low 32 = index (stable ties)
__global__ void k_keys(const float* __restrict__ scores, unsigned long long* __restrict__ keys) {
  int b = blockIdx.y;
  int i = blockIdx.x * blockDim.x + threadIdx.x;   // 0..PSORT-1
  unsigned long long key = ~0ULL;                  // padding sorts last (ascending sort)
  if (i < NANCH) {
    int a = i % NA;
    int pix = i / NA;                              // y*FW + x
    float s = scores[((size_t)b * (2*NA) + (NA + a)) * (FH*FW) + pix];
    unsigned u = __float_as_uint(s);
    u ^= (u >> 31) ? 0xFFFFFFFFu : 0x80000000u;    // monotone ascending
    u = ~u;                                        // descending
    key = ((unsigned long long)u << 32) | (unsigned)i;
  }
  keys[(size_t)b * PSORT + i] = key;
}

// bbox_transform_inv + clip for every anchor
__global__ void k_props(const float* __restrict__ deltas, const float* __restrict__ im_info,
                        float* __restrict__ props) {
  int b = blockIdx.y;
  int i = blockIdx.x * blockDim.x + threadIdx.x;   // 0..NANCH-1
  int a = i % NA;
  int pix = i / NA;
  int x = pix % FW, y = pix / FW;
  float sx = x * 16.0f, sy = y * 16.0f;
  float ax1 = kBase[a][0] + sx, ay1 = kBase[a][1] + sy;
  float ax2 = kBase[a][2] + sx, ay2 = kBase[a][3] + sy;
  float w = ax2 - ax1 + 1.0f, h = ay2 - ay1 + 1.0f;
  float ctx = ax1 + 0.5f * w, cty = ay1 + 0.5f * h;
  const float* dl = deltas + (size_t)b * (4*NA) * (FH*FW);
  const int s4 = FH*FW;
  float dx = dl[(4*a + 0) * s4 + pix];
  float dy = dl[(4*a + 1) * s4 + pix];
  float dw = dl[(4*a + 2) * s4 + pix];
  float dh = dl[(4*a + 3) * s4 + pix];
  float pcx = dx * w + ctx, pcy = dy * h + cty;
  float pw = expf(dw) * w,  ph = expf(dh) * h;
  float x1 = pcx - 0.5f * pw, y1 = pcy - 0.5f * ph;
  float x2 = pcx + 0.5f * pw, y2 = pcy + 0.5f * ph;
  float hm = im_info[b*3 + 0] - 1.0f;
  float wm = im_info[b*3 + 1] - 1.0f;
  x1 = fminf(fmaxf(x1, 0.0f), wm);
  y1 = fminf(fmaxf(y1, 0.0f), hm);
  x2 = fminf(fmaxf(x2, 0.0f), wm);
  y2 = fminf(fmaxf(y2, 0.0f), hm);
  float* p = props + ((size_t)b * NANCH + i) * 4;
  p[0] = x1; p[1] = y1; p[2] = x2; p[3] = y2;
}

// ---- bitonic sort (ascending, per batch slab of PSORT) ----
__global__ void k_sort_local(unsigned long long* __restrict__ keys) {
  __shared__ unsigned long long sh[2048];
  unsigned long long* kb = keys + (size_t)blockIdx.y * PSORT;
  int base = blockIdx.x * 2048;
  int t = threadIdx.x;                      // 0..1023
  sh[t] = kb[base + t];
  sh[t + 1024] = kb[base + t + 1024];
  __syncthreads();
  for (int k = 2; k <= 2048; k <<= 1) {
    for (int j = k >> 1; j >= 1; j >>= 1) {
      int i = 2*t - (t & (j - 1));
      bool dir = (((base + i) & k) == 0);
      unsigned long long xa = sh[i], xb = sh[i + j];
      if ((xa > xb) == dir) { sh[i] = xb; sh[i + j] = xa; }
      __syncthreads();
    }
  }
  kb[base + t] = sh[t];
  kb[base + t + 1024] = sh[t + 1024];
}

__global__ void k_sort_global(unsigned long long* __restrict__ keys, int j, int k) {
  unsigned long long* kb = keys + (size_t)blockIdx.y * PSORT;
  int t = blockIdx.x * blockDim.x + threadIdx.x;   // 0..PSORT/2-1
  int i = 2*t - (t & (j - 1));
  bool dir = ((i & k) == 0);
  unsigned long long xa = kb[i], xb = kb[i + j];
  if ((xa > xb) == dir) { kb[i] = xb; kb[i + j] = xa; }
}

__global__ void k_sort_merge_local(unsigned long long* __restrict__ keys, int k) {
  __shared__ unsigned long long sh[2048];
  unsigned long long* kb = keys + (size_t)blockIdx.y * PSORT;
  int base = blockIdx.x * 2048;
  int t = threadIdx.x;
  sh[t] = kb[base + t];
  sh[t + 1024] = kb[base + t + 1024];
  __syncthreads();
  for (int j = 1024; j >= 1; j >>= 1) {
    int i = 2*t - (t & (j - 1));
    bool dir = (((base + i) & k) == 0);
    unsigned long long xa = sh[i], xb = sh[i + j];
    if ((xa > xb) == dir) { sh[i] = xb; sh[i + j] = xa; }
    __syncthreads();
  }
  kb[base + t] = sh[t];
  kb[base + t + 1024] = sh[t + 1024];
}

__global__ void k_gather(const unsigned long long* __restrict__ keys,
                         const float* __restrict__ props, float* __restrict__ boxes) {
  int b = blockIdx.y;
  int r = blockIdx.x * blockDim.x + threadIdx.x;
  if (r >= PRE) return;
  unsigned idx = (unsigned)(keys[(size_t)b * PSORT + r] & 0xFFFFFFFFu);
  const float* p = props + ((size_t)b * NANCH + idx) * 4;
  float* q = boxes + ((size_t)b * PRE + r) * 4;
  q[0] = p[0]; q[1] = p[1]; q[2] = p[2]; q[3] = p[3];
}

// Suppression bit-matrix. One wave handles a 16-row x 32-col tile (= one mask word per row).
// IoU denominator area[M]+area[N]-inter computed on the matrix pipe:
//   D = A(16x4) * B(4x16) + C, A=[rowArea,1,0,0], B=[1;colArea;0;0], C=-inter.
// WMMA C/D layout: VGPR k, lane l -> (M = k + 8*(l>=16), N = l&15)  [CDNA5 ISA]
__global__ void k_nms_mask(const float* __restrict__ boxes, unsigned* __restrict__ mask) {
  int wave = (blockIdx.x * blockDim.x + threadIdx.x) >> 5;   // grid sized exactly
  int lane = threadIdx.x & 31;
  int rt = wave / MASKW;            // row tile 0..374
  int ct = wave % MASKW;            // 32-col tile 0..187
  int halfsel = lane >> 4;          // 0: lanes 0-15, 1: lanes 16-31
  int ln = lane & 15;

  // row boxes: lane l holds row rt*16 + (l&15) for the shfl broadcasts
  int rrow = rt * 16 + ln;
  float rx1 = boxes[rrow*4 + 0], ry1 = boxes[rrow*4 + 1];
  float rx2 = boxes[rrow*4 + 2], ry2 = boxes[rrow*4 + 3];
  float rArea = (rx2 - rx1 + 1.0f) * (ry2 - ry1 + 1.0f);

  // two 16-col subtiles
  float cx1[2], cy1[2], cx2[2], cy2[2], cAr[2];
  bool cval[2];
  #pragma unroll
  for (int s = 0; s < 2; s++) {
    int c = ct*32 + s*16 + ln;
    cval[s] = (c < PRE);
    int cc = cval[s] ? c : (PRE - 1);
    cx1[s] = boxes[cc*4 + 0]; cy1[s] = boxes[cc*4 + 1];
    cx2[s] = boxes[cc*4 + 2]; cy2[s] = boxes[cc*4 + 3];
    cAr[s] = (cx2[s] - cx1[s] + 1.0f) * (cy2[s] - cy1[s] + 1.0f);
  }

  // A-matrix 16x4: lanes 0-15 hold K=0,1 ; lanes 16-31 hold K=2,3 (zero rows)
  v2f A;
  A.x = halfsel ? 0.0f : rArea;   // A[M][0] = rowArea[M]
  A.y = halfsel ? 0.0f : 1.0f;    // A[M][1] = 1
  // B-matrix 4x16: VGPR0 = rows K=0 (lanes0-15) / K=2 (lanes16-31); VGPR1 = K=1 / K=3
  v2f B0, B1;
  B0.x = halfsel ? 0.0f : 1.0f;      B0.y = halfsel ? 0.0f : cAr[0];
  B1.x = halfsel ? 0.0f : 1.0f;      B1.y = halfsel ? 0.0f : cAr[1];

  // intersections per (k, lane) into negated C
  v8f c0, c1;
  #pragma unroll
  for (int k = 0; k < 8; k++) {
    int M = k + 8 * halfsel;
    float qx1 = __shfl(rx1, M, 32);
    float qy1 = __shfl(ry1, M, 32);
    float qx2 = __shfl(rx2, M, 32);
    float qy2 = __shfl(ry2, M, 32);
    #pragma unroll
    for (int s = 0; s < 2; s++) {
      float ix1 = fmaxf(qx1, cx1[s]);
      float iy1 = fmaxf(qy1, cy1[s]);
      float ix2 = fminf(qx2, cx2[s]);
      float iy2 = fminf(qy2, cy2[s]);
      float iw = fmaxf(ix2 - ix1 + 1.0f, 0.0f);
      float ih = fmaxf(iy2 - iy1 + 1.0f, 0.0f);
      float inter = iw * ih;
      if (s == 0) c0[k] = -inter; else c1[k] = -inter;
    }
  }

  // denominators on the matrix pipe: d = rowArea + colArea - inter
  v8f d0 = __builtin_amdgcn_wmma_f32_16x16x4_f32(false, A, false, B0, (short)0, c0, false, false);
  v8f d1 = __builtin_amdgcn_wmma_f32_16x16x4_f32(false, A, false, B1, (short)0, c1, false, false);

  #pragma unroll
  for (int k = 0; k < 8; k++) {
    float i0 = -c0[k], i1 = -c1[k];
    bool b0 = (i0 > NMS_T * d0[k]) && cval[0];
    bool b1 = (i1 > NMS_T * d1[k]) && cval[1];
    unsigned long long bal0 = __ballot(b0);
    unsigned long long bal1 = __ballot(b1);
    unsigned lo0 = (unsigned)(bal0 & 0xFFFFu),  lo1 = (unsigned)(bal1 & 0xFFFFu);
    unsigned hi0 = (unsigned)((bal0 >> 16) & 0xFFFFu), hi1 = (unsigned)((bal1 >> 16) & 0xFFFFu);
    unsigned wlo = lo0 | (lo1 << 16);   // row k,   cols ct*32..+31
    unsigned whi = hi0 | (hi1 << 16);   // row k+8
    if (lane == 0) {
      mask[(size_t)(rt*16 + k)     * MASKW + ct] = wlo;
      mask[(size_t)(rt*16 + k + 8) * MASKW + ct] = whi;
    }
  }
}

// Greedy sweep: one wave. "removed" bitset (6144 bits) lives in registers, word w owned by
// lane (w&31), slot (w>>5). All control flow is wave-uniform.
__global__ void k_nms_sweep(const float* __restrict__ boxes, const unsigned* __restrict__ mask,
                            float* __restrict__ out, int b) {
  int lane = threadIdx.x;
  unsigned rm0 = 0, rm1 = 0, rm2 = 0, rm3 = 0, rm4 = 0, rm5 = 0;
  int count = 0;
  for (int i = 0; i < PRE; i++) {
    int w = i >> 5;
    unsigned word;
    switch (w >> 5) {
      case 0: word = rm0; break;
      case 1: word = rm1; break;
      case 2: word = rm2; break;
      case 3: word = rm3; break;
      case 4: word = rm4; break;
      default: word = rm5; break;
    }
    unsigned sup = (unsigned)__shfl((int)word, w & 31, 32);
    if (!((sup >> (i & 31)) & 1u)) {
      if (lane == 0) {
        const float* bp = boxes + (size_t)i * 4;
        float* op = out + ((size_t)b * POST + count) * 5;
        op[0] = (float)b;
        op[1] = bp[0]; op[2] = bp[1]; op[3] = bp[2]; op[4] = bp[3];
      }
      count++;
      if (count >= POST) break;
      const unsigned* mrow = mask + (size_t)i * MASKW;
      int wi;
      wi = lane;        if (wi < MASKW) rm0 |= mrow[wi];
      wi = lane + 32;   if (wi < MASKW) rm1 |= mrow[wi];
      wi = lane + 64;   if (wi < MASKW) rm2 |= mrow[wi];
      wi = lane + 96;   if (wi < MASKW) rm3 |= mrow[wi];
      wi = lane + 128;  if (wi < MASKW) rm4 |= mrow[wi];
      wi = lane + 160;  if (wi < MASKW) rm5 |= mrow[wi];
    }
  }
}

extern "C" void kernel_launch(void* const* d_in, const int* in_sizes, int n_in,
                              void* d_out, int out_size, void* d_ws, size_t ws_size,
                              hipStream_t stream) {
  (void)in_sizes; (void)n_in; (void)ws_size;
  const float* scores  = (const float*)d_in[0];
  const float* deltas  = (const float*)d_in[1];
  const float* im_info = (const float*)d_in[2];
  float* out = (float*)d_out;

  char* ws = (char*)d_ws;
  unsigned long long* keys = (unsigned long long*)ws;                       // 4*65536*8  = 2 MB
  size_t off = (size_t)NB * PSORT * sizeof(unsigned long long);
  float* props = (float*)(ws + off);                                        // 4*36864*16 = 2.36 MB
  off += (size_t)NB * NANCH * 4 * sizeof(float);
  float* boxes = (float*)(ws + off);                                        // 4*6000*16  = 384 KB
  off += (size_t)NB * PRE * 4 * sizeof(float);
  unsigned* mask = (unsigned*)(ws + off);                                   // 6000*188*4 = 4.5 MB (reused per batch)

  // 1) zero output (poisoned by harness; rows past #kept must be 0)
  k_zero<<<(out_size + 255) / 256, 256, 0, stream>>>(out, out_size);
  // 2) sort keys + 3) decoded proposals
  k_keys <<<dim3(PSORT / 256, NB), 256, 0, stream>>>(scores, keys);
  k_props<<<dim3(NANCH / 256, NB), 256, 0, stream>>>(deltas, im_info, props);
  // 4) bitonic sort, 21 launches total (j<=1024 fused in LDS)
  k_sort_local<<<dim3(PSORT / 2048, NB), 1024, 0, stream>>>(keys);
  for (int k = 4096; k <= PSORT; k <<= 1) {
    for (int j = k >> 1; j >= 2048; j >>= 1)
      k_sort_global<<<dim3((PSORT / 2) / 256, NB), 256, 0, stream>>>(keys, j, k);
    k_sort_merge_local<<<dim3(PSORT / 2048, NB), 1024, 0, stream>>>(keys, k);
  }
  // 5) top-6000 boxes per batch
  k_gather<<<dim3((PRE + 255) / 256, NB), 256, 0, stream>>>(keys, props, boxes);
  // 6) per-batch mask (WMMA) + sweep; mask buffer reused (stream-ordered)
  const int waves = (PRE / 16) * MASKW;   // 375*188 = 70500, divisible by 4 waves/block
  for (int b = 0; b < NB; b++) {
    const float* bb = boxes + (size_t)b * PRE * 4;
    k_nms_mask <<<waves / 4, 128, 0, stream>>>(bb, mask);
    k_nms_sweep<<<1, 32, 0, stream>>>(bb, mask, out, b);
  }
}